// GateBase_6828998000758
// MI455X (gfx1250) — compile-verified
//
#include <hip/hip_runtime.h>

// ---------------------------------------------------------------------------
// 2-qubit complex gate on a 25-qubit fp32 state, targets (10,3).
//   axis 10 -> element stride 2^14 ; axis 3 -> element stride 2^21
// Memory-bound: 512 MB @ 23.3 TB/s ~ 22 us. Math done via
// V_WMMA_F32_16X16X4_F32: per 16 groups, D(16x16) = A(16x4)*B(4x16)+C twice
// (K=8 real-embedded complex matvec [[Mr,-Mi],[Mi,Mr]]).
//   A = state data  (A-layout per ISA: reg v, lane l: M=l&15, K=v+2*(l>=16))
//   B = W^T (4x16, cols>=8 zero), per-lane built from the 4x4 gate matrix
//   D = reg v, lane l: group=v+8*(l>=16), component=l&15  -> LDS transpose
//       -> fully coalesced 128B streaming stores.
// ---------------------------------------------------------------------------

typedef __attribute__((ext_vector_type(2))) float v2f;
typedef __attribute__((ext_vector_type(8))) float v8f;

#define BIT_A 14u            // target axis 10 stride bit
#define BIT_B 21u            // target axis 3  stride bit
#define N_STATE  (1u << 25)
#define N_GROUPS (1u << 23)
#define GROUPS_PER_CHUNK 32u
#define N_CHUNKS (N_GROUPS / GROUPS_PER_CHUNK)   // 2^18

// Insert zero bits at positions 14 and 21 of a 23-bit group index.
__device__ __forceinline__ unsigned expand23(unsigned g) {
    unsigned lo  = g & 0x3FFFu;          // bits 0..13 -> 0..13
    unsigned mid = (g >> 14) & 0x3Fu;    // bits 14..19 -> 15..20
    unsigned hi  = (g >> 20) & 0x7u;     // bits 20..22 -> 22..24
    return lo | (mid << 15) | (hi << 22);
}

__global__ __launch_bounds__(256) void gate2q_wmma(
    const float* __restrict__ qr, const float* __restrict__ qi,
    const float* __restrict__ mr, const float* __restrict__ mi,
    float* __restrict__ outr, float* __restrict__ outi)
{
    // Wave-private LDS tile: 32 groups x 8 comps, stride 9 (bank-conflict free)
    __shared__ float lds[8][GROUPS_PER_CHUNK * 9];
    const unsigned tid   = threadIdx.x;
    const unsigned lane  = tid & 31u;
    const unsigned wslot = tid >> 5;
    float* wl = lds[wslot];

    const unsigned l15 = lane & 15u;
    const unsigned h   = lane >> 4;              // 0: lanes 0-15, 1: lanes 16-31

    // ---- Build B operands (W^T, padded 4x16) once per thread -------------
    // B reg v, this lane: K = v + 2h, N = l15. Zero for N >= 8.
    v2f B1 = {0.f, 0.f}, B2 = {0.f, 0.f};
    if (l15 < 8u) {
        const unsigned n = l15;
        #pragma unroll
        for (int v = 0; v < 2; ++v) {
            const unsigned k = (unsigned)v + 2u * h;   // input component j
            float b1, b2;
            if (n < 4u) {                              // out_r rows of W
                b1 =  mr[n * 4u + k];
                b2 = -mi[n * 4u + k];
            } else {                                   // out_i rows of W
                b1 =  mi[(n - 4u) * 4u + k];
                b2 =  mr[(n - 4u) * 4u + k];
            }
            if (v == 0) { B1.x = b1; B2.x = b2; }
            else        { B1.y = b1; B2.y = b2; }
        }
    }

    const unsigned wavesPerBlk = blockDim.x >> 5;
    const unsigned waveGid = blockIdx.x * wavesPerBlk + wslot;
    const unsigned nWaves  = gridDim.x * wavesPerBlk;   // divides N_CHUNKS by launch config

    for (unsigned c = waveGid; c < N_CHUNKS; c += nWaves) {
        const unsigned eb = expand23(c * GROUPS_PER_CHUNK);

        // ---- two 16-group sub-chunks: load A, 2x WMMA, D -> LDS ----------
        #pragma unroll
        for (unsigned s = 0; s < 2u; ++s) {
            // A layout: group = l15, K = v + 2h; component j offset:
            //   j=2h   -> +(h<<14) ; j=2h+1 -> +(h<<14)+(1<<21)
            const unsigned ebs = eb + s * 16u + l15 + (h << BIT_A);
            v2f Ar, Ai;
            Ar.x = __builtin_nontemporal_load(qr + ebs);
            Ar.y = __builtin_nontemporal_load(qr + ebs + (1u << BIT_B));
            Ai.x = __builtin_nontemporal_load(qi + ebs);
            Ai.y = __builtin_nontemporal_load(qi + ebs + (1u << BIT_B));

            v8f acc = {};
            acc = __builtin_amdgcn_wmma_f32_16x16x4_f32(
                      false, Ar, false, B1, (short)0, acc, false, false);
            acc = __builtin_amdgcn_wmma_f32_16x16x4_f32(
                      false, Ai, false, B2, (short)0, acc, false, false);

            // D reg v, lane: group = v + 8h (+16s), comp = l15 (comps >=8 are 0)
            if (l15 < 8u) {
                const unsigned gbase = 8u * h + 16u * s;
                #pragma unroll
                for (int v = 0; v < 8; ++v)
                    wl[(gbase + (unsigned)v) * 9u + l15] = acc[v];
            }
        }
        __syncthreads();   // uniform trips (launch guarantees N_CHUNKS % nWaves == 0)

        // ---- LDS -> coalesced streaming stores: comp n, group = lane -----
        const unsigned ebl = eb + lane;
        #pragma unroll
        for (unsigned n = 0; n < 8u; ++n) {
            const float val = wl[lane * 9u + n];
            const unsigned off = (((n & 3u) >> 1) << BIT_A) | ((n & 1u) << BIT_B);
            if (n < 4u) __builtin_nontemporal_store(val, outr + ebl + off);
            else        __builtin_nontemporal_store(val, outi + ebl + off);
        }
        __syncthreads();
    }
}

extern "C" void kernel_launch(void* const* d_in, const int* in_sizes, int n_in,
                              void* d_out, int out_size, void* d_ws, size_t ws_size,
                              hipStream_t stream) {
    const float* qr = (const float*)d_in[0];   // qs_real  (2^25)
    const float* qi = (const float*)d_in[1];   // qs_imag  (2^25)
    const float* mr = (const float*)d_in[2];   // mat_real (16)
    const float* mi = (const float*)d_in[3];   // mat_imag (16)
    float* outr = (float*)d_out;               // tuple concat: real then imag
    float* outi = outr + N_STATE;

    // 2048 blocks * 8 waves = 16384 waves; N_CHUNKS (2^18) / 16384 = 16 exactly,
    // so every wave (and every block) runs the same trip count (__syncthreads safe).
    dim3 grid(2048), block(256);
    hipLaunchKernelGGL(gate2q_wmma, grid, block, 0, stream,
                       qr, qi, mr, mi, outr, outi);
}